// CPUBackboneWrapper_51539608361
// MI455X (gfx1250) — compile-verified
//
#include <hip/hip_runtime.h>
#include <hip/hip_bf16.h>

#define NPTS   16384
#define KNB    20
#define SLOPE  0.2f
#define NEGINF (-3.402823e38f)
#define KNN_WAVES 4

typedef float v2f __attribute__((ext_vector_type(2)));
typedef float v8f __attribute__((ext_vector_type(8)));

__device__ __forceinline__ v8f wmma_f32_k4(v2f a, v2f b, v8f c) {
  // D = A(16x4) * B(4x16) + C, full fp32 (matches reference numerics)
  return __builtin_amdgcn_wmma_f32_16x16x4_f32(false, a, false, b, (short)0, c, false, false);
}

__device__ __forceinline__ float leaky(float y) {
  return fmaxf(y, 0.0f) + SLOPE * fminf(y, 0.0f);
}

// ---------------------------------------------------------------------------
// x0 = [coords.T ; feats.T]  (4 x N, channel-major)
__global__ void pack_kernel(const float* __restrict__ coords, const float* __restrict__ feats,
                            float* __restrict__ x0) {
  int n = blockIdx.x * blockDim.x + threadIdx.x;
  x0[0 * NPTS + n] = coords[n * 3 + 0];
  x0[1 * NPTS + n] = coords[n * 3 + 1];
  x0[2 * NPTS + n] = coords[n * 3 + 2];
  x0[3 * NPTS + n] = feats[n];
}

// xx[n] = sum_c x[c,n]^2
__global__ void colnorm_kernel(const float* __restrict__ x, int C, float* __restrict__ xx) {
  int n = blockIdx.x * blockDim.x + threadIdx.x;
  float s = 0.0f;
  for (int c = 0; c < C; ++c) { float v = x[c * NPTS + n]; s += v * v; }
  xx[n] = s;
}

// ---------------------------------------------------------------------------
// Fused WMMA Gram + streaming top-20.  Each wave owns a 16-row tile and scans
// all 1024 column tiles.  A fragments are hoisted into registers (templated C)
// so the inner loop is pure: coalesced global B loads -> chained WMMA.
template <int C>
__global__ void __launch_bounds__(KNN_WAVES * 32)
knn_kernel(const float* __restrict__ x, const float* __restrict__ xx,
           int* __restrict__ nbidx) {
  __shared__ float ScrAll[KNN_WAVES][16 * 80];   // D-tile bounce (256) / merge (1280)

  const int lane = threadIdx.x & 31;
  const int wave = threadIdx.x >> 5;
  const int row0 = (blockIdx.x * KNN_WAVES + wave) * 16;
  float* DT = ScrAll[wave];

  const int sel0 = (lane < 16) ? 0 : 2;
  const int sel1 = sel0 + 1;
  const int ln   = lane & 15;
  const int mbase = (lane < 16) ? 0 : 8;

  // hoisted A fragments: afrag[cb] = A[m=ln][k = 4cb + sel0/sel1] = x[k][row0+ln]
  v2f afrag[C / 4];
  #pragma unroll
  for (int cb = 0; cb < C / 4; ++cb) {
    afrag[cb].x = x[(cb * 4 + sel0) * NPTS + row0 + ln];
    afrag[cb].y = x[(cb * 4 + sel1) * NPTS + row0 + ln];
  }

  const int r  = lane >> 1;          // row handled in top-k phase
  const int h  = lane & 1;           // column half (8 cols each)
  const float xxrow = xx[row0 + r];

  float vals[KNB]; int ids[KNB];
  #pragma unroll
  for (int s = 0; s < KNB; ++s) { vals[s] = NEGINF; ids[s] = 0; }
  float minval = NEGINF; int minslot = 0;

  for (int jt = 0; jt < NPTS / 16; ++jt) {
    const int j0 = jt * 16;
    __builtin_prefetch(xx + j0 + 2048 + ln, 0, 1);   // global_prefetch_b8 (stays in ws)
    v8f acc = {};
    #pragma unroll
    for (int cb = 0; cb < C / 4; ++cb) {
      v2f b;
      b.x = x[(cb * 4 + sel0) * NPTS + j0 + ln];
      b.y = x[(cb * 4 + sel1) * NPTS + j0 + ln];
      acc = wmma_f32_k4(afrag[cb], b, acc);
    }
    #pragma unroll
    for (int p = 0; p < 8; ++p) DT[(mbase + p) * 16 + ln] = acc[p];  // D row = mbase+p

    const float4 q0 = *reinterpret_cast<const float4*>(xx + j0 + h * 8);
    const float4 q1 = *reinterpret_cast<const float4*>(xx + j0 + h * 8 + 4);
    float xcs[8] = {q0.x, q0.y, q0.z, q0.w, q1.x, q1.y, q1.z, q1.w};

    #pragma unroll
    for (int k2 = 0; k2 < 8; ++k2) {
      const float g  = DT[r * 16 + h * 8 + k2];
      const float nd = 2.0f * g - xxrow - xcs[k2];      // neg squared distance
      if (nd > minval) {
        const int jidx = j0 + h * 8 + k2;
        #pragma unroll
        for (int s = 0; s < KNB; ++s) if (s == minslot) { vals[s] = nd; ids[s] = jidx; }
        minval = vals[0]; minslot = 0;
        #pragma unroll
        for (int s = 1; s < KNB; ++s) if (vals[s] < minval) { minval = vals[s]; minslot = s; }
      }
    }
  }

  // merge the two half-row lists (reuse scratch: 80 dwords per row)
  #pragma unroll
  for (int s = 0; s < KNB; ++s) {
    DT[r * 80 + h * 20 + s] = vals[s];
    ((int*)DT)[r * 80 + 40 + h * 20 + s] = ids[s];
  }
  if (h == 0) {   // lane 2r selects global top-20 of the 40 candidates
    for (int t = 0; t < KNB; ++t) {
      float best = NEGINF; int bs = 0;
      #pragma unroll
      for (int s = 0; s < 40; ++s) {
        float vv = DT[r * 80 + s];
        if (vv > best) { best = vv; bs = s; }
      }
      nbidx[(row0 + r) * KNB + t] = ((int*)DT)[r * 80 + 40 + bs];
      DT[r * 80 + bs] = NEGINF;
    }
  }
}

// ---------------------------------------------------------------------------
// Wp = [A ; B-A] from w = [A | B] (O x 2C): edgeconv collapse weights
__global__ void buildwp_kernel(const float* __restrict__ w, int O, int C, float* __restrict__ Wp) {
  int t = blockIdx.x * blockDim.x + threadIdx.x;
  if (t >= O * C) return;
  int o = t / C, c = t % C;
  float a = w[o * 2 * C + c];
  float b = w[o * 2 * C + C + c];
  Wp[o * C + c]       = a;
  Wp[(O + o) * C + c] = b - a;
}

// ---------------------------------------------------------------------------
// Generic fp32 WMMA GEMM: Y(MxN) = A(MxK,row-major) * B(KxN,row-major)
// mode 0: raw   1: leaky(y*s+c)   2: y*s+c   3: transposed store y+c (sem head)
__global__ void __launch_bounds__(256)
gemm_kernel(int M, int K, int N, int Mreal,
            const float* __restrict__ A, const float* __restrict__ B, float* __restrict__ Y,
            int mode, const float* __restrict__ sv, const float* __restrict__ cv, int ldT) {
  __shared__ float As[512 * 16];   // As[k][m], K <= 512
  const int tid = threadIdx.x;
  const int m0  = blockIdx.y * 16;
  for (int t = tid; t < K * 16; t += 256) {
    int ck = t % K, i = t / K;
    As[ck * 16 + i] = (m0 + i < Mreal) ? A[(m0 + i) * K + ck] : 0.0f;
  }
  __syncthreads();

  const int lane = tid & 31, wave = tid >> 5;
  const int j0   = (blockIdx.x * 8 + wave) * 16;
  const int sel0 = (lane < 16) ? 0 : 2, sel1 = sel0 + 1, ln = lane & 15;
  const int mbase = (lane < 16) ? 0 : 8;

  v8f acc = {};
  for (int cb = 0; cb < K; cb += 4) {
    v2f a, b;
    a.x = As[(cb + sel0) * 16 + ln];
    a.y = As[(cb + sel1) * 16 + ln];
    b.x = B[(cb + sel0) * N + j0 + ln];
    b.y = B[(cb + sel1) * N + j0 + ln];
    acc = wmma_f32_k4(a, b, acc);
  }
  #pragma unroll
  for (int p = 0; p < 8; ++p) {
    const int m = m0 + mbase + p;
    const int col = j0 + ln;
    float y = acc[p];
    if (mode == 1) {
      Y[m * N + col] = leaky(y * sv[m] + cv[m]);
    } else if (mode == 2) {
      Y[m * N + col] = y * sv[m] + cv[m];
    } else if (mode == 3) {
      if (m < Mreal) Y[col * ldT + m] = y + cv[m];
    } else {
      Y[m * N + col] = y;
    }
  }
}

// ---------------------------------------------------------------------------
// x_out[o,n] = max_k leaky(s[o]*(v[o,idx[n,k]] + u[o,n]) + c[o]);  v is L2-resident
__global__ void gathermax_kernel(const float* __restrict__ v, const float* __restrict__ u,
                                 const int* __restrict__ nbidx, const float* __restrict__ sv,
                                 const float* __restrict__ cv, float* __restrict__ out) {
  int t = blockIdx.x * blockDim.x + threadIdx.x;
  int n = t & (NPTS - 1);
  int o = t >> 14;
  const float uo = u[o * NPTS + n];
  const float so = sv[o], co = cv[o];
  const int* ip = nbidx + n * KNB;
  float acc = NEGINF;
  #pragma unroll 4
  for (int k = 0; k < KNB; ++k) {
    int j = ip[k];
    acc = fmaxf(acc, leaky(so * (v[o * NPTS + j] + uo) + co));
  }
  out[o * NPTS + n] = acc;
}

// ms_coords broadcast + ms_masks zeros
__global__ void tail_kernel(const float* __restrict__ coords, float* __restrict__ outc,
                            float* __restrict__ outm) {
  int t = blockIdx.x * blockDim.x + threadIdx.x;
  outc[t] = coords[t % (3 * NPTS)];     // 3 x (N x 3)
  if (t < 3 * NPTS) outm[t] = 0.0f;     // masks = false
}

// ---------------------------------------------------------------------------
extern "C" void kernel_launch(void* const* d_in, const int* in_sizes, int n_in,
                              void* d_out, int out_size, void* d_ws, size_t ws_size,
                              hipStream_t stream) {
  (void)in_sizes; (void)n_in; (void)out_size; (void)ws_size;
  const int N = NPTS;
  const float* coords = (const float*)d_in[0];
  const float* feats  = (const float*)d_in[1];
  const float* w1 = (const float*)d_in[2],  *s1 = (const float*)d_in[3],  *c1 = (const float*)d_in[4];
  const float* w2 = (const float*)d_in[5],  *s2 = (const float*)d_in[6],  *c2 = (const float*)d_in[7];
  const float* w3 = (const float*)d_in[8],  *s3 = (const float*)d_in[9],  *c3 = (const float*)d_in[10];
  const float* w4 = (const float*)d_in[11], *s4 = (const float*)d_in[12], *c4 = (const float*)d_in[13];
  const float* w5 = (const float*)d_in[14], *s5 = (const float*)d_in[15], *c5 = (const float*)d_in[16];
  const float* wf = (const float*)d_in[17], *sf = (const float*)d_in[18], *cf = (const float*)d_in[19];
  const float* wsem = (const float*)d_in[20], *bsem = (const float*)d_in[21];

  // workspace layout (fp32 unless noted)
  float* ws    = (float*)d_ws;
  float* x0v   = ws;                         // 4N
  float* xx    = x0v + 4 * N;                // N
  int*   nbidx = (int*)(xx + N);             // 20N ints
  float* Wp    = (float*)(nbidx + 20 * N);   // <= 512*128
  float* VU    = Wp + 512 * 128;             // 512N   ( [v ; u] )
  float* cat   = VU + 512 * N;               // 512N   ( x1|x2|x3|x4 )
  float* x5    = cat + 512 * N;              // 512N

  // output layout: ms_feats(3*256*N) | ms_coords(3*N*3) | ms_masks(3*N) | sem(N*20)
  float* out        = (float*)d_out;
  float* out_feats  = out;
  float* out_coords = out + 768 * N;
  float* out_masks  = out_coords + 9 * N;
  float* out_sem    = out_masks + 3 * N;

  pack_kernel<<<N / 256, 256, 0, stream>>>(coords, feats, x0v);

  auto launch_knn = [&](const float* xin, int C) {
    dim3 g(N / (16 * KNN_WAVES));
    if (C == 4)       knn_kernel<4>  <<<g, KNN_WAVES * 32, 0, stream>>>(xin, xx, nbidx);
    else if (C == 64) knn_kernel<64> <<<g, KNN_WAVES * 32, 0, stream>>>(xin, xx, nbidx);
    else              knn_kernel<128><<<g, KNN_WAVES * 32, 0, stream>>>(xin, xx, nbidx);
  };

  auto run_layer = [&](const float* xin, int C, int O, const float* w, const float* s,
                       const float* c, float* xout) {
    colnorm_kernel<<<N / 256, 256, 0, stream>>>(xin, C, xx);
    launch_knn(xin, C);
    buildwp_kernel<<<(O * C + 255) / 256, 256, 0, stream>>>(w, O, C, Wp);
    gemm_kernel<<<dim3(N / 128, (2 * O) / 16), 256, 0, stream>>>(
        2 * O, C, N, 2 * O, Wp, xin, VU, 0, nullptr, nullptr, 0);
    gathermax_kernel<<<(O * N) / 256, 256, 0, stream>>>(VU, VU + O * N, nbidx, s, c, xout);
  };

  run_layer(x0v,           4,   64,  w1, s1, c1, cat);             // x1
  run_layer(cat,           64,  64,  w2, s2, c2, cat + 64 * N);    // x2
  run_layer(cat + 64 * N,  64,  128, w3, s3, c3, cat + 128 * N);   // x3
  run_layer(cat + 128 * N, 128, 256, w4, s4, c4, cat + 256 * N);   // x4

  // x5 = leaky(w5 @ cat * s5 + c5)
  gemm_kernel<<<dim3(N / 128, 512 / 16), 256, 0, stream>>>(
      512, 512, N, 512, w5, cat, x5, 1, s5, c5, 0);
  // ms_feats = wf @ x5 * sf + cf   (one 768x512 GEMM, written straight to d_out)
  gemm_kernel<<<dim3(N / 128, 768 / 16), 256, 0, stream>>>(
      768, 512, N, 768, wf, x5, out_feats, 2, sf, cf, 0);
  // sem_logits = (ms_feats[2].T @ wsem.T + bsem), stored (N,20); M padded to 32
  gemm_kernel<<<dim3(N / 128, 2), 256, 0, stream>>>(
      32, 256, N, 20, wsem, out_feats + 512 * N, out_sem, 3, nullptr, bsem, 20);

  tail_kernel<<<(9 * N) / 256, 256, 0, stream>>>(coords, out_coords, out_masks);
}